// TopDownAttention_20151986553272
// MI455X (gfx1250) — compile-verified
//
#include <hip/hip_runtime.h>
#include <hip/hip_bf16.h>

// MI455X / gfx1250, wave32. Fused top-down attention:
//   tkey = tf @ Wt^T + bt          (kernel 1, -> d_ws, needs 4000*64*4 = 1MB)
//   out  = cf + tf[c2t] * sigmoid( tanh(tkey[c2t] + cf@Wc^T + bc) @ aw )
// Both GEMMs use V_WMMA_F32_16X16X4_F32 (full fp32 precision; compute stays
// below the 23.3 TB/s HBM roofline of ~44us for the 1GB of cf/out traffic).

typedef float v2f __attribute__((ext_vector_type(2)));
typedef float v4f __attribute__((ext_vector_type(4)));
typedef float v8f __attribute__((ext_vector_type(8)));

#define CELL_DIM 128
#define HIDDEN   64
#define LDS_W_STRIDE 132   // 128 + 4 pad -> strided float2 LDS reads hit distinct banks

// Stage W[64 x 128] (row-major) into padded LDS [64 x 132]. Block-cooperative.
__device__ __forceinline__ void stage_w_lds(float* ldsW, const float* __restrict__ W) {
    for (int i = threadIdx.x; i < (HIDDEN * CELL_DIM) / 4; i += blockDim.x) {
        int r = i >> 5;            // 32 float4 per 128-wide row
        int c = (i & 31) << 2;
        v4f v = *(const v4f*)(W + r * CELL_DIM + c);
        *(v4f*)(&ldsW[r * LDS_W_STRIDE + c]) = v;
    }
}

// One wave computes Q[64 x 16] = W[64x128] @ X_tile[16x128]^T via 128 WMMAs.
//   A (16x4 per step): A[m,k] = W[mt*16 + m, k]   (W staged in LDS, padded)
//   B (4x16 per step): B[k,n] = X[base + n, k]    (held in 64 VGPRs)
// A/B lane layout (32-bit, wave32): lane l -> row/col l%16,
//   VGPR0 = K = 4*ko + 2*(l/16), VGPR1 = K+1.
// D layout: lane l, vgpr g = D[M = g + 8*(l/16), N = l%16].
__device__ __forceinline__ void wave_gemm_q(const float* __restrict__ X, long base,
                                            const float* ldsW, v8f acc[4]) {
    const int lane = threadIdx.x & 31;
    const int half = lane >> 4;
    const int n16  = lane & 15;

    const float* xrow = X + (size_t)(base + n16) * CELL_DIM + 2 * half;
    v2f b[32];
#pragma unroll
    for (int ko = 0; ko < 32; ++ko)
        b[ko] = *(const v2f*)(xrow + 4 * ko);

#pragma unroll
    for (int mt = 0; mt < 4; ++mt) {
        v8f c = {};
        const float* wrow = ldsW + (mt * 16 + n16) * LDS_W_STRIDE + 2 * half;
#pragma unroll
        for (int ko = 0; ko < 32; ++ko) {
            v2f a = *(const v2f*)(wrow + 4 * ko);
            // 8 args: (neg_a, A, neg_b, B, c_mod, C, reuse_a, reuse_b)
            c = __builtin_amdgcn_wmma_f32_16x16x4_f32(
                    false, a, false, b[ko], (short)0, c, false, false);
        }
        acc[mt] = c;
    }
}

// Kernel 1: tkey[n_tissue, 64] = tf @ Wt^T + bt
__global__ void __launch_bounds__(128)
tissue_key_kernel(const float* __restrict__ tf, const float* __restrict__ Wt,
                  const float* __restrict__ bt, float* __restrict__ tkey,
                  int n_tissue) {
    __shared__ float ldsW[HIDDEN * LDS_W_STRIDE];
    stage_w_lds(ldsW, Wt);
    __syncthreads();

    int wave  = (int)((blockIdx.x * blockDim.x + threadIdx.x) >> 5);
    int tiles = n_tissue >> 4;
    if (wave >= tiles) return;

    long base = (long)wave * 16;
    v8f acc[4];
    wave_gemm_q(tf, base, ldsW, acc);

    const int lane = threadIdx.x & 31, half = lane >> 4, n16 = lane & 15;
    long row = base + n16;
#pragma unroll
    for (int mt = 0; mt < 4; ++mt)
#pragma unroll
        for (int g = 0; g < 8; ++g) {
            int h = mt * 16 + g + 8 * half;
            tkey[row * HIDDEN + h] = acc[mt][g] + bt[h];
        }
}

// Kernel 2: fused query-GEMM + gather + tanh/dot/sigmoid + residual write.
// 16 cells per wave, 4 waves per block -> 64 cells/block (1M % 64 == 0).
__global__ void __launch_bounds__(128)
fused_cell_kernel(const float* __restrict__ cf, const float* __restrict__ tf,
                  const int* __restrict__ c2t, const float* __restrict__ Wc,
                  const float* __restrict__ bc, const float* __restrict__ aw,
                  const float* __restrict__ tkey, float* __restrict__ out,
                  int n_cell) {
    __shared__ float ldsW[HIDDEN * LDS_W_STRIDE];
    __shared__ float ldsGate[4 * 16];
    stage_w_lds(ldsW, Wc);
    __syncthreads();

    const int  wslot = threadIdx.x >> 5;
    const long gwave = (long)blockIdx.x * (blockDim.x >> 5) + wslot;
    const long base  = gwave * 16;
    const int  lane  = threadIdx.x & 31, half = lane >> 4, n16 = lane & 15;
    const bool active = base < (long)n_cell;

    if (active) {
        v8f acc[4];
        wave_gemm_q(cf, base, ldsW, acc);

        const long cell = base + n16;
        const int  t    = c2t[cell];
        const float* tk = tkey + (size_t)t * HIDDEN;

        float s = 0.f;
#pragma unroll
        for (int mt = 0; mt < 4; ++mt)
#pragma unroll
            for (int g = 0; g < 8; ++g) {
                int h = mt * 16 + g + 8 * half;
                float v = acc[mt][g] + bc[h] + tk[h];
                s += tanhf(v) * aw[h];
            }
        // lanes l and l+16 each hold half of the 64 hidden terms for cell l%16
        s += __shfl_xor(s, 16, 32);
        float gate = 1.0f / (1.0f + __expf(-s));
        if (half == 0) ldsGate[wslot * 16 + n16] = gate;
    }
    __syncthreads();

    if (active) {
        const int col = lane * 4;   // 32 lanes * float4 = one full 512B row
#pragma unroll 4
        for (int r = 0; r < 16; ++r) {
            long row = base + r;
            int  t   = c2t[row];
            float g  = ldsGate[wslot * 16 + r];
            v4f cv = *(const v4f*)(cf + (size_t)row * CELL_DIM + col);
            v4f tv = *(const v4f*)(tf + (size_t)t   * CELL_DIM + col);
            v4f o  = cv + tv * g;
            *(v4f*)(out + (size_t)row * CELL_DIM + col) = o;
        }
    }
}

extern "C" void kernel_launch(void* const* d_in, const int* in_sizes, int n_in,
                              void* d_out, int out_size, void* d_ws, size_t ws_size,
                              hipStream_t stream) {
    const float* cf  = (const float*)d_in[0];
    const float* tf  = (const float*)d_in[1];
    const int*   c2t = (const int*)  d_in[2];
    const float* Wt  = (const float*)d_in[3];
    const float* bt  = (const float*)d_in[4];
    const float* Wc  = (const float*)d_in[5];
    const float* bc  = (const float*)d_in[6];
    const float* aw  = (const float*)d_in[7];
    float* outp = (float*)d_out;
    float* tkey = (float*)d_ws;     // needs n_tissue * 64 * 4 = 1 MB

    const int n_cell   = in_sizes[2];
    const int n_tissue = in_sizes[1] / CELL_DIM;

    // Kernel 1: 250 wave-tiles, 4 waves/block
    int tiles_t = n_tissue >> 4;
    int blocks1 = (tiles_t + 3) / 4;
    tissue_key_kernel<<<blocks1, 128, 0, stream>>>(tf, Wt, bt, tkey, n_tissue);

    // Kernel 2: 62500 wave-tiles, 4 waves/block -> 15625 blocks (exact)
    int tiles_c = n_cell >> 4;
    int blocks2 = (tiles_c + 3) / 4;
    fused_cell_kernel<<<blocks2, 128, 0, stream>>>(cf, tf, c2t, Wc, bc, aw,
                                                   tkey, outp, n_cell);
}